// Partition_4879082848867
// MI455X (gfx1250) — compile-verified
//
#include <hip/hip_runtime.h>
#include <stdint.h>

#define THREADS 256
#define ITEMS   16
#define TILE    (THREADS * ITEMS)   // 4096 elements per block

typedef unsigned int uint32x4 __attribute__((ext_vector_type(4)));
typedef int          int32x4  __attribute__((ext_vector_type(4)));
typedef int          int32x8  __attribute__((ext_vector_type(8)));

// Block-wide exclusive scan of one int per thread (wave32: warpSize==32 on
// gfx1250). Returns exclusive prefix; writes block total to *total_out.
__device__ __forceinline__ int block_exscan(int v, int tid, int* total_out) {
    const int lane = tid & 31;
    const int wid  = tid >> 5;
    int x = v;
#pragma unroll
    for (int d = 1; d < 32; d <<= 1) {
        int y = __shfl_up(x, (unsigned)d, 32);
        if (lane >= d) x += y;
    }
    __shared__ int wsum[THREADS / 32];
    if (lane == 31) wsum[wid] = x;
    __syncthreads();
    if (wid == 0) {
        int w = (lane < THREADS / 32) ? wsum[lane] : 0;
#pragma unroll
        for (int d = 1; d < THREADS / 32; d <<= 1) {
            int y = __shfl_up(w, (unsigned)d, 32);
            if (lane >= d) w += y;
        }
        if (lane < THREADS / 32) wsum[lane] = w;
    }
    __syncthreads();
    int waveoff = (wid > 0) ? wsum[wid - 1] : 0;
    *total_out  = wsum[THREADS / 32 - 1];
    return waveoff + x - v;   // exclusive prefix for this thread
}

// Pass 1: per-block count of (id == 0)
__global__ void count_zeros_kernel(const int* __restrict__ ids, int n,
                                   int* __restrict__ blockSums) {
    const int tid       = threadIdx.x;
    const int blockBase = blockIdx.x * TILE;
    const int base      = blockBase + tid * ITEMS;
    int s = 0;
    if (blockBase + TILE <= n) {
        const int4* p = (const int4*)(ids + base);
        int4 a[4];
#pragma unroll
        for (int q = 0; q < 4; ++q) a[q] = p[q];
#pragma unroll
        for (int q = 0; q < 4; ++q) {
            s += (a[q].x == 0) ? 1 : 0;
            s += (a[q].y == 0) ? 1 : 0;
            s += (a[q].z == 0) ? 1 : 0;
            s += (a[q].w == 0) ? 1 : 0;
        }
    } else {
        for (int j = 0; j < ITEMS; ++j) {
            int i = base + j;
            if (i < n) s += (ids[i] == 0) ? 1 : 0;
        }
    }
    int total;
    (void)block_exscan(s, tid, &total);
    if (tid == 0) blockSums[blockIdx.x] = total;
}

// Pass 2: single-block exclusive scan over up to 4096 block sums.
__global__ void scan_blocks_kernel(int* __restrict__ blockSums, int numBlocks,
                                   int* __restrict__ totalOut) {
    const int tid  = threadIdx.x;
    const int base = tid * ITEMS;
    int v[ITEMS];
    int s = 0;
#pragma unroll
    for (int j = 0; j < ITEMS; ++j) {
        v[j] = (base + j < numBlocks) ? blockSums[base + j] : 0;
        s += v[j];
    }
    int total;
    int run = block_exscan(s, tid, &total);
#pragma unroll
    for (int j = 0; j < ITEMS; ++j) {
        if (base + j < numBlocks) blockSums[base + j] = run;
        run += v[j];
    }
    if (tid == 0) totalOut[0] = total;
}

// Pass 3: TDM-stage the ids tile into LDS, compute stable ranks, scatter.
__global__ void scatter_kernel(const int*    __restrict__ ids,
                               const float4* __restrict__ vals,
                               const float*  __restrict__ keys,
                               int n,
                               const int* __restrict__ blockOffsets,
                               const int* __restrict__ totalPtr,
                               float4* __restrict__ outVals,
                               float*  __restrict__ outKeys,
                               int*    __restrict__ outIdx,
                               int*    __restrict__ outCounts) {
    const int tid       = threadIdx.x;
    const int blockBase = blockIdx.x * TILE;
    const int base      = blockBase + tid * ITEMS;
    const int count0    = *totalPtr;               // uniform scalar load
    const bool full     = (blockBase + TILE <= n);

    __shared__ __align__(16) int tileIds[TILE];    // 16 KB of the WGP's 320 KB LDS

    // gfx1250 global_prefetch_b8: start pulling payload lines early.
    if (base < n) {
        __builtin_prefetch(&vals[base], 0, 0);
        __builtin_prefetch(&keys[base], 0, 0);
    }

    int myid[ITEMS];
    int s = 0;
    if (full) {
#if __has_builtin(__builtin_amdgcn_tensor_load_to_lds)
        // --- Tensor Data Mover: DMA the 4096-int ids tile into LDS. ---
        if (tid < 32) {   // one wave issues the descriptor
            unsigned long long gaddr =
                (unsigned long long)(uintptr_t)(ids + blockBase);
            // Flat->LDS offset: LDS aperture maps addr[31:0] to the LDS byte
            // address (CDNA5 aperture rule), so low 32 bits are the offset.
            unsigned int ldsOff =
                (unsigned int)(uintptr_t)(&tileIds[0]);

            uint32x4 g0;
            g0[0] = 1u;                                    // count=1 (valid D#)
            g0[1] = ldsOff;                                // lds_addr (bytes)
            g0[2] = (unsigned int)(gaddr & 0xFFFFFFFFu);   // global_addr[31:0]
            g0[3] = (unsigned int)((gaddr >> 32) & 0x01FFFFFFu)
                    | (2u << 30);                          // addr[56:32] | type=2

            int32x8 g1;
            g1[0] = 0x00020000;            // workgroup_mask=0, data_size=2 (4B)
            g1[1] = (int)((unsigned)TILE << 16);   // tensor_dim0 = 4096 (lo16)
            g1[2] = 0x00010000;            // tensor_dim0 hi=0, tensor_dim1=1
            g1[3] = (int)((unsigned)TILE << 16);   // tile_dim0 = 4096
            g1[4] = 1;                     // tile_dim1=1, tile_dim2=0
            g1[5] = TILE;                  // tensor_dim0_stride = 4096
            g1[6] = 0;
            g1[7] = 0;

            int32x4 gz = {0, 0, 0, 0};
#if defined(__clang_major__) && (__clang_major__ >= 23)
            int32x8 gz8 = {0, 0, 0, 0, 0, 0, 0, 0};
            __builtin_amdgcn_tensor_load_to_lds(g0, g1, gz, gz, gz8, 0);
#else
            __builtin_amdgcn_tensor_load_to_lds(g0, g1, gz, gz, 0);
#endif
            __builtin_amdgcn_s_wait_tensorcnt(0);          // s_wait_tensorcnt 0
        }
        __syncthreads();                                   // LDS visible to all waves
        {
            const int4* lp = (const int4*)(&tileIds[tid * ITEMS]);
            int4 a[4];
#pragma unroll
            for (int q = 0; q < 4; ++q) a[q] = lp[q];      // ds_load_b128 x4
#pragma unroll
            for (int q = 0; q < 4; ++q) {
                myid[4 * q + 0] = a[q].x;
                myid[4 * q + 1] = a[q].y;
                myid[4 * q + 2] = a[q].z;
                myid[4 * q + 3] = a[q].w;
            }
        }
#else
        // Fallback: clause'd global b128 loads.
        const int4* p = (const int4*)(ids + base);
        int4 a[4];
#pragma unroll
        for (int q = 0; q < 4; ++q) a[q] = p[q];
#pragma unroll
        for (int q = 0; q < 4; ++q) {
            myid[4 * q + 0] = a[q].x;
            myid[4 * q + 1] = a[q].y;
            myid[4 * q + 2] = a[q].z;
            myid[4 * q + 3] = a[q].w;
        }
#endif
#pragma unroll
        for (int j = 0; j < ITEMS; ++j) s += (myid[j] == 0) ? 1 : 0;
    } else {
        for (int j = 0; j < ITEMS; ++j) {
            int i   = base + j;
            int p   = (i < n) ? ids[i] : 1;        // OOB -> partition 1, never stored
            myid[j] = p;
            s += (p == 0) ? 1 : 0;
        }
    }

    int total;
    int ex      = block_exscan(s, tid, &total);
    int running = blockOffsets[blockIdx.x] + ex;   // ex0 at this thread's first element

    if (full) {
        // Batch all payload loads (16x b128 vals + 4x b128 keys), then stores.
        float4 v[ITEMS];
        float  k[ITEMS];
        const float4* kp = (const float4*)(keys + base);
        float4 kv[4];
#pragma unroll
        for (int q = 0; q < 4; ++q) kv[q] = kp[q];
#pragma unroll
        for (int j = 0; j < ITEMS; ++j) v[j] = vals[base + j];
#pragma unroll
        for (int q = 0; q < 4; ++q) {
            k[4 * q + 0] = kv[q].x;
            k[4 * q + 1] = kv[q].y;
            k[4 * q + 2] = kv[q].z;
            k[4 * q + 3] = kv[q].w;
        }
#pragma unroll
        for (int j = 0; j < ITEMS; ++j) {
            int i = base + j;
            int dest;
            if (myid[j] == 0) {
                dest = running;                    // stable rank among zeros
                ++running;
            } else {
                dest = count0 + i - running;       // stable rank among ones
            }
            outVals[dest] = v[j];                  // global_store_b128
            outKeys[dest] = k[j];
            outIdx[dest]  = i;
        }
    } else {
        for (int j = 0; j < ITEMS; ++j) {
            int i = base + j;
            if (i < n) {
                int dest;
                if (myid[j] == 0) {
                    dest = running;
                    ++running;
                } else {
                    dest = count0 + i - running;
                }
                outVals[dest] = vals[i];
                outKeys[dest] = keys[i];
                outIdx[dest]  = i;
            }
        }
    }

    if (blockBase == 0 && tid == 0) {
        outCounts[0] = count0;
        outCounts[1] = n - count0;
    }
}

extern "C" void kernel_launch(void* const* d_in, const int* in_sizes, int n_in,
                              void* d_out, int out_size, void* d_ws, size_t ws_size,
                              hipStream_t stream) {
    (void)n_in; (void)out_size; (void)ws_size;

    const int*    ids  = (const int*)d_in[0];       // partition_ids [N] int32
    const float4* vals = (const float4*)d_in[1];    // input_vals [N,4] f32
    const float*  keys = (const float*)d_in[2];     // input_keys [N] f32

    const int n         = in_sizes[0];
    const int numBlocks = (n + TILE - 1) / TILE;    // 4096 for N = 16,777,216

    // Workspace: [0 .. numBlocks) block sums, then 1 int for count0.
    int* blockSums = (int*)d_ws;
    int* totalPtr  = blockSums + numBlocks;

    // Output layout (f32 buffer; ints stored bitwise):
    // vals [4N] | keys [N] | idx [N] | counts [2]
    float*  out     = (float*)d_out;
    float4* outVals = (float4*)out;
    float*  outKeys = out + (size_t)4 * n;
    int*    outIdx  = (int*)(out + (size_t)5 * n);
    int*    outCnt  = (int*)(out + (size_t)6 * n);

    count_zeros_kernel<<<numBlocks, THREADS, 0, stream>>>(ids, n, blockSums);
    scan_blocks_kernel<<<1, THREADS, 0, stream>>>(blockSums, numBlocks, totalPtr);
    scatter_kernel<<<numBlocks, THREADS, 0, stream>>>(ids, vals, keys, n,
                                                      blockSums, totalPtr,
                                                      outVals, outKeys, outIdx, outCnt);
}